// CoattentiveLayer_46995532152849
// MI455X (gfx1250) — compile-verified
//
#include <hip/hip_runtime.h>
#include <cmath>

#define B_   64
#define LC_  1024
#define LQ_  128
#define D_   1024
#define LC1_ 1025
#define LQ1_ 129

typedef float v2f __attribute__((ext_vector_type(2)));
typedef float v8f __attribute__((ext_vector_type(8)));

#define BMT 64
#define BNT 64
#define BKT 32

// Row-view helper: MODE 0 = row-major, MODE 1 = virtual concat (row 0 lives at `extra`)
template<int MODE>
__device__ __forceinline__ const float* rowPtr(const float* base, long ld,
                                               const float* extra, int r) {
  if constexpr (MODE == 1) {
    return (r == 0) ? extra : (base + (long)(r - 1) * ld);
  } else {
    return base + (long)r * ld;
  }
}

// Generic f32 WMMA GEMM:  C[m,n] = sum_k A[m,k]*B[k,n]  (per batch = blockIdx.z)
// AM: 0 row-major, 1 concat(sentinel row0), 2 transposed (A[m,k]=base[k*ld+m])
// BM: 0 row-major, 1 concat rows,           2 transposed (B[k,n]=base[n*ld+k])
// CM: 0 row-major, 1 split (m==0 -> Cex, else base+(m-1)*ld), 2 skip row 0
// Software-pipelined: tile k+1 is fetched from global into registers while
// tile k is computed from LDS.
template<int AM, int BMo, int CM, bool TANH>
__global__ __launch_bounds__(256)
void gemm_f32_wmma(const float* __restrict__ A, long ldA, long sA,
                   const float* __restrict__ Aex, long sAex,
                   const float* __restrict__ Bp, long ldB, long sB,
                   const float* __restrict__ Bex, long sBex,
                   float* __restrict__ C, long ldC, long sC,
                   float* __restrict__ Cex, long sCex,
                   const float* __restrict__ bias,
                   int M, int N, int K)
{
  __shared__ float As[BMT][BKT + 1];
  __shared__ float Bs[BKT][BNT + 1];

  const int tid = threadIdx.x;
  const int bz  = blockIdx.z;
  const int m0  = blockIdx.y * BMT;
  const int n0  = blockIdx.x * BNT;

  const float* Ab   = A  + (long)bz * sA;
  const float* Aexb = Aex ? (Aex + (long)bz * sAex) : nullptr;
  const float* Bb   = Bp + (long)bz * sB;
  const float* Bexb = Bex ? (Bex + (long)bz * sBex) : nullptr;
  float* Cb   = C + (long)bz * sC;
  float* Cexb = Cex ? (Cex + (long)bz * sCex) : nullptr;

  const int w    = tid >> 5;
  const int lane = tid & 31;
  const int wm0  = (w & 3) * 16;    // wave's 16-row strip
  const int wn0  = (w >> 2) * 32;   // wave's 32-col strip (2 wmma tiles)

  v8f acc0 = {0.f,0.f,0.f,0.f,0.f,0.f,0.f,0.f};
  v8f acc1 = {0.f,0.f,0.f,0.f,0.f,0.f,0.f,0.f};

  // WMMA f32 16x16x4 fragment addressing (wave32):
  // A: lane<16 holds row=lane K={k,k+1}; lane>=16 holds row=lane-16 K={k+2,k+3}
  const int arow = wm0 + (lane & 15);
  const int kb   = (lane >> 4) << 1;
  const int bcol = wn0 + (lane & 15);

  // staging registers (only one A path / one B path is live per instantiation)
  float  rA[8];  float4 rA4[2];
  float  rB[8];  float4 rB4[2];

  auto loadAtile = [&](int k0) {
    if constexpr (AM == 2) {
      #pragma unroll
      for (int i = 0; i < 8; ++i) {
        int idx = tid + i * 256;
        int kk = idx >> 6, mm = idx & 63;       // coalesced along m
        int gk = k0 + kk, gm = m0 + mm;
        rA[i] = (gk < K && gm < M) ? Ab[(long)gk * ldA + gm] : 0.f;
      }
    } else if constexpr (AM == 1) {             // rows contiguous, ld%4==0 -> float4
      #pragma unroll
      for (int i = 0; i < 2; ++i) {
        int idx = tid + i * 256;
        int mm = idx >> 3, k4 = (idx & 7) << 2;
        int gm = m0 + mm, gk = k0 + k4;
        float4 v = make_float4(0.f, 0.f, 0.f, 0.f);
        if (gm < M && gk + 3 < K) {
          const float* rp = rowPtr<1>(Ab, ldA, Aexb, gm);
          v = *(const float4*)(rp + gk);
        }
        rA4[i] = v;
      }
    } else {                                    // row-major, possibly odd ld (129)
      #pragma unroll
      for (int i = 0; i < 8; ++i) {
        int idx = tid + i * 256;
        int mm = idx >> 5, kk = idx & 31;
        int gm = m0 + mm, gk = k0 + kk;
        rA[i] = (gm < M && gk < K) ? Ab[(long)gm * ldA + gk] : 0.f;
      }
    }
  };
  auto storeAtile = [&]() {
    if constexpr (AM == 2) {
      #pragma unroll
      for (int i = 0; i < 8; ++i) {
        int idx = tid + i * 256;
        As[idx & 63][idx >> 6] = rA[i];
      }
    } else if constexpr (AM == 1) {
      #pragma unroll
      for (int i = 0; i < 2; ++i) {
        int idx = tid + i * 256;
        int mm = idx >> 3, k4 = (idx & 7) << 2;
        As[mm][k4 + 0] = rA4[i].x; As[mm][k4 + 1] = rA4[i].y;
        As[mm][k4 + 2] = rA4[i].z; As[mm][k4 + 3] = rA4[i].w;
      }
    } else {
      #pragma unroll
      for (int i = 0; i < 8; ++i) {
        int idx = tid + i * 256;
        As[idx >> 5][idx & 31] = rA[i];
      }
    }
  };
  auto loadBtile = [&](int k0) {
    if constexpr (BMo == 2) {
      #pragma unroll
      for (int i = 0; i < 8; ++i) {
        int idx = tid + i * 256;
        int nn = idx >> 5, kk = idx & 31;       // coalesced along k
        int gn = n0 + nn, gk = k0 + kk;
        rB[i] = (gn < N && gk < K) ? Bb[(long)gn * ldB + gk] : 0.f;
      }
    } else {                                    // rows contiguous along n -> float4
      #pragma unroll
      for (int i = 0; i < 2; ++i) {
        int idx = tid + i * 256;
        int kk = idx >> 4, n4 = (idx & 15) << 2;
        int gk = k0 + kk, gn = n0 + n4;
        float4 v = make_float4(0.f, 0.f, 0.f, 0.f);
        if (gk < K && gn + 3 < N) {
          const float* rp = rowPtr<BMo>(Bb, ldB, Bexb, gk);
          v = *(const float4*)(rp + gn);
        }
        rB4[i] = v;
      }
    }
  };
  auto storeBtile = [&]() {
    if constexpr (BMo == 2) {
      #pragma unroll
      for (int i = 0; i < 8; ++i) {
        int idx = tid + i * 256;
        Bs[idx & 31][idx >> 5] = rB[i];
      }
    } else {
      #pragma unroll
      for (int i = 0; i < 2; ++i) {
        int idx = tid + i * 256;
        int kk = idx >> 4, n4 = (idx & 15) << 2;
        Bs[kk][n4 + 0] = rB4[i].x; Bs[kk][n4 + 1] = rB4[i].y;
        Bs[kk][n4 + 2] = rB4[i].z; Bs[kk][n4 + 3] = rB4[i].w;
      }
    }
  };

  // ---- software pipeline ----
  loadAtile(0);
  loadBtile(0);
  for (int k0 = 0; k0 < K; k0 += BKT) {
    storeAtile();
    storeBtile();
    __syncthreads();

    const int kn = k0 + BKT;
    if (kn < K) {            // issue next tile's global loads; overlap with WMMA
      loadAtile(kn);
      loadBtile(kn);
    }

    #pragma unroll
    for (int kk = 0; kk < BKT; kk += 4) {
      v2f a, b0, b1;
      a.x  = As[arow][kk + kb];       a.y  = As[arow][kk + kb + 1];
      b0.x = Bs[kk + kb][bcol];       b0.y = Bs[kk + kb + 1][bcol];
      b1.x = Bs[kk + kb][bcol + 16];  b1.y = Bs[kk + kb + 1][bcol + 16];
      acc0 = __builtin_amdgcn_wmma_f32_16x16x4_f32(false, a, false, b0,
                                                   (short)0, acc0, false, false);
      acc1 = __builtin_amdgcn_wmma_f32_16x16x4_f32(false, a, false, b1,
                                                   (short)0, acc1, false, false);
    }
    __syncthreads();
  }

  // ---- store: C/D layout -> VGPR i: lanes 0-15 row=i, lanes 16-31 row=i+8 ----
  const int rhalf = (lane >> 4) * 8;
  const int ccol  = lane & 15;
  #pragma unroll
  for (int i = 0; i < 8; ++i) {
    int gm = m0 + wm0 + rhalf + i;
    if (gm >= M) continue;
    #pragma unroll
    for (int t = 0; t < 2; ++t) {
      int gn = n0 + wn0 + ccol + t * 16;
      if (gn >= N) continue;
      float v = (t == 0) ? acc0[i] : acc1[i];
      if constexpr (TANH) v = tanhf(v + bias[gn]);
      if constexpr (CM == 0) {
        Cb[(long)gm * ldC + gn] = v;
      } else if constexpr (CM == 1) {
        if (gm == 0) Cexb[gn] = v;
        else         Cb[(long)(gm - 1) * ldC + gn] = v;
      } else {
        if (gm > 0)  Cb[(long)(gm - 1) * ldC + gn] = v;
      }
    }
  }
}

// Softmax over the context dim (columns of affinity[c][qi]).
// Block = 16x16, each block handles 16 qi-columns for one batch.
__global__ __launch_bounds__(256)
void softmax_ctx_kernel(const float* __restrict__ aff,
                        const unsigned char* __restrict__ cpad,
                        float* __restrict__ Ac)
{
  __shared__ float red[16][17];
  const int tx = threadIdx.x & 15, ty = threadIdx.x >> 4;
  const int b = blockIdx.y;
  const int col = blockIdx.x * 16 + tx;
  const bool colok = col < LQ1_;
  const float* ab = aff + (long)b * LC1_ * LQ1_;
  const unsigned char* pb = cpad + (long)b * LC_;

  float mx = -3.4e38f;
  for (int r = ty; r < LC1_; r += 16) {
    float v = colok ? ab[(long)r * LQ1_ + col] : -3.4e38f;
    if (r > 0 && pb[r - 1]) v = -1e10f;
    mx = fmaxf(mx, v);
  }
  red[ty][tx] = mx; __syncthreads();
  if (ty == 0) { float m = red[0][tx];
    #pragma unroll
    for (int i = 1; i < 16; ++i) m = fmaxf(m, red[i][tx]);
    red[0][tx] = m; }
  __syncthreads();
  mx = red[0][tx];
  __syncthreads();

  float s = 0.f;
  for (int r = ty; r < LC1_; r += 16) {
    if (colok) {
      float v = ab[(long)r * LQ1_ + col];
      if (r > 0 && pb[r - 1]) v = -1e10f;
      s += __expf(v - mx);
    }
  }
  red[ty][tx] = s; __syncthreads();
  if (ty == 0) { float t = 0.f;
    #pragma unroll
    for (int i = 0; i < 16; ++i) t += red[i][tx];
    red[0][tx] = t; }
  __syncthreads();
  const float inv = 1.f / red[0][tx];

  float* ob = Ac + (long)b * LC1_ * LQ1_;
  for (int r = ty; r < LC1_; r += 16) {
    if (colok) {
      float v = ab[(long)r * LQ1_ + col];
      if (r > 0 && pb[r - 1]) v = -1e10f;
      ob[(long)r * LQ1_ + col] = __expf(v - mx) * inv;
    }
  }
}

// Softmax over the question dim (rows of affinity[c][qi]); one wave32 per row c.
__global__ __launch_bounds__(256)
void softmax_q_kernel(const float* __restrict__ aff,
                      const unsigned char* __restrict__ qpad,
                      float* __restrict__ Aq)
{
  const int b = blockIdx.y;
  const int w = threadIdx.x >> 5, lane = threadIdx.x & 31;
  const int c = blockIdx.x * 8 + w;
  if (c >= LC1_) return;
  const float* row = aff + (long)b * LC1_ * LQ1_ + (long)c * LQ1_;
  const unsigned char* pb = qpad + (long)b * LQ_;

  float vals[5];
  float mx = -3.4e38f;
  #pragma unroll
  for (int j = 0; j < 5; ++j) {
    int qi = lane + j * 32;
    float v = -3.4e38f;
    if (qi < LQ1_) {
      v = row[qi];
      if (qi > 0 && pb[qi - 1]) v = -1e10f;
    }
    vals[j] = v;
    mx = fmaxf(mx, v);
  }
  #pragma unroll
  for (int off = 16; off; off >>= 1) mx = fmaxf(mx, __shfl_xor(mx, off, 32));

  float es[5]; float s = 0.f;
  #pragma unroll
  for (int j = 0; j < 5; ++j) {
    int qi = lane + j * 32;
    float e = (qi < LQ1_) ? __expf(vals[j] - mx) : 0.f;
    es[j] = e; s += e;
  }
  #pragma unroll
  for (int off = 16; off; off >>= 1) s += __shfl_xor(s, off, 32);
  const float inv = 1.f / s;

  float* orow = Aq + (long)b * LC1_ * LQ1_ + (long)c * LQ1_;
  #pragma unroll
  for (int j = 0; j < 5; ++j) {
    int qi = lane + j * 32;
    if (qi < LQ1_) orow[qi] = es[j] * inv;
  }
}

extern "C" void kernel_launch(void* const* d_in, const int* in_sizes, int n_in,
                              void* d_out, int out_size, void* d_ws, size_t ws_size,
                              hipStream_t stream)
{
  (void)in_sizes; (void)n_in; (void)out_size; (void)ws_size;
  const float* context  = (const float*)d_in[0];
  const float* question = (const float*)d_in[1];
  const unsigned char* cpad = (const unsigned char*)d_in[2];
  const unsigned char* qpad = (const unsigned char*)d_in[3];
  const float* projW    = (const float*)d_in[4];
  const float* projb    = (const float*)d_in[5];
  const float* sentinel = (const float*)d_in[6];

  float* outC = (float*)d_out;                               // [B, LC, 2D]
  float* outQ = outC + (size_t)B_ * LC_ * 2 * D_;            // [B, LQ, 2D]

  float* ws   = (float*)d_ws;
  float* qbuf = ws;                                          // [B,129,1024]
  float* aff  = qbuf + (size_t)B_ * LQ1_ * D_;               // [B,1025,129]
  float* Ac   = aff  + (size_t)B_ * LC1_ * LQ1_;             // [B,1025,129]
  float* Aq   = Ac   + (size_t)B_ * LC1_ * LQ1_;             // [B,1025,129] ([c][qi])
  float* Sc0  = Aq   + (size_t)B_ * LC1_ * LQ1_;             // [B,1024] S_c row 0
  float* Sq0  = Sc0  + (size_t)B_ * D_;                      // [B,1024] S_q row 0

  dim3 blk(256);

  // 1) q = tanh([sentinel1; question] @ W^T + b)          M=129 N=1024 K=1024
  gemm_f32_wmma<1,2,0,true><<<dim3(D_/BNT, (LQ1_+BMT-1)/BMT, B_), blk, 0, stream>>>(
      question, D_, (long)LQ_*D_, sentinel + D_, 0,
      projW,    D_, 0,           nullptr, 0,
      qbuf,     D_, (long)LQ1_*D_, nullptr, 0,
      projb, LQ1_, D_, D_);

  // 2) affinity[c][qi] = [sentinel0; ctx] @ q^T           M=1025 N=129 K=1024
  gemm_f32_wmma<1,2,0,false><<<dim3((LQ1_+BNT-1)/BNT, (LC1_+BMT-1)/BMT, B_), blk, 0, stream>>>(
      context, D_, (long)LC_*D_, sentinel, 0,
      qbuf,    D_, (long)LQ1_*D_, nullptr, 0,
      aff,  LQ1_, (long)LC1_*LQ1_, nullptr, 0,
      nullptr, LC1_, LQ1_, D_);

  // 3) softmaxes
  softmax_ctx_kernel<<<dim3((LQ1_+15)/16, B_), blk, 0, stream>>>(aff, cpad, Ac);
  softmax_q_kernel  <<<dim3((LC1_+7)/8,  B_), blk, 0, stream>>>(aff, qpad, Aq);

  // 4) S_c = A_c^T @ ctxv : row0 -> Sc0, rows 1.. -> outQ[:,:,1024:]   M=129 N=1024 K=1025
  gemm_f32_wmma<2,1,1,false><<<dim3(D_/BNT, (LQ1_+BMT-1)/BMT, B_), blk, 0, stream>>>(
      Ac, LQ1_, (long)LC1_*LQ1_, nullptr, 0,
      context, D_, (long)LC_*D_, sentinel, 0,
      outQ + D_, 2L*D_, (long)LQ_*2*D_, Sc0, D_,
      nullptr, LQ1_, D_, LC1_);

  // 5) S_q = Aq[c][qi] @ q : row0 -> Sq0, rows 1.. -> outC[:,:,1024:]  M=1025 N=1024 K=129
  gemm_f32_wmma<0,0,1,false><<<dim3(D_/BNT, (LC1_+BMT-1)/BMT, B_), blk, 0, stream>>>(
      Aq, LQ1_, (long)LC1_*LQ1_, nullptr, 0,
      qbuf, D_, (long)LQ1_*D_, nullptr, 0,
      outC + D_, 2L*D_, (long)LC_*2*D_, Sq0, D_,
      nullptr, LC1_, D_, LQ1_);

  // 6) coattn_context = Aq[c][qi] @ S_c : rows 1.. -> outC[:,:,0:1024] M=1025 N=1024 K=129
  gemm_f32_wmma<0,1,2,false><<<dim3(D_/BNT, (LC1_+BMT-1)/BMT, B_), blk, 0, stream>>>(
      Aq, LQ1_, (long)LC1_*LQ1_, nullptr, 0,
      outQ + D_, 2L*D_, (long)LQ_*2*D_, Sc0, D_,
      outC, 2L*D_, (long)LC_*2*D_, nullptr, 0,
      nullptr, LC1_, D_, LQ1_);

  // 7) coattn_question = A_c^T @ S_q : rows 1.. -> outQ[:,:,0:1024]    M=129 N=1024 K=1025
  gemm_f32_wmma<2,1,2,false><<<dim3(D_/BNT, (LQ1_+BMT-1)/BMT, B_), blk, 0, stream>>>(
      Ac, LQ1_, (long)LC1_*LQ1_, nullptr, 0,
      outC + D_, 2L*D_, (long)LC_*2*D_, Sq0, D_,
      outQ, 2L*D_, (long)LQ_*2*D_, nullptr, 0,
      nullptr, LQ1_, D_, LC1_);
}